// superglue_3393024163968
// MI455X (gfx1250) — compile-verified
//
#include <hip/hip_runtime.h>
#include <hip/hip_bf16.h>

// SuperGlue matching loss, gather-reduce formulation.
// B=16, N=M=2048, scores [B, N+1, M+1] f32. Output: scalar f32 mean loss.
//
// Key identity: loss_b = -(tp_b + tn_b) / (xx_b + M), so per block we reduce
// only (s = tp+tn partial, x = count partial).
//
// Wave reduction uses V_WMMA_F32_16X16X4_F32 (f32 in, f32 out -> exact):
//   A (16x4 f32, 2 VGPRs/lane):  lane l<16: {A[l][0], A[l][1]},
//                                lane l>=16: {A[l-16][2], A[l-16][3]}
//   B = all-ones (layout independent), C = 0.
//   => D[i][j] = A[i][0..3] row sum = v_i + v_{i+16}  (we zero the 2nd VGPR)
//   D (16x16 f32, 8 VGPRs): lane j<16 holds D[0..7][j], lane j>=16 holds D[8..15][j-16]
//   => sum of 8 accumulators + one xor-16 shuffle = full 32-lane sum.

typedef __attribute__((ext_vector_type(2))) float v2f;
typedef __attribute__((ext_vector_type(8))) float v8f;

__device__ __forceinline__ float wave_sum_wmma(float v) {
    v2f a;      a.x = v;   a.y = 0.0f;   // A: lane value in K-slot 0 (or 2), zero in 1 (or 3)
    v2f bones;  bones.x = 1.0f; bones.y = 1.0f;  // B: all ones -> layout independent
    v8f c = {};
    // 8 args: (neg_a, A, neg_b, B, c_mod, C, reuse_a, reuse_b)
    c = __builtin_amdgcn_wmma_f32_16x16x4_f32(
        /*neg_a=*/false, a, /*neg_b=*/false, bones,
        /*c_mod=*/(short)0, c, /*reuse_a=*/false, /*reuse_b=*/false);
    float t = c[0] + c[1] + c[2] + c[3] + c[4] + c[5] + c[6] + c[7];
    // t(lane<16)  = sum_{i=0..7}  (v_i + v_{i+16})
    // t(lane>=16) = sum_{i=8..15} (v_i + v_{i+16})
    t += __shfl_xor(t, 16, 32);
    return t;   // every lane: sum over all 32 lanes
}

// One block per (batch b, slice). 256 threads = 8 waves. Each thread handles
// one row-gather (tp) and one dustbin-row element (tn/xx).
__global__ void __launch_bounds__(256)
superglue_partial_kernel(const int* __restrict__ gt0,
                         const int* __restrict__ gt1,
                         const float* __restrict__ scores,
                         float* __restrict__ ws,
                         int n, int m, int slices) {
    const int b     = blockIdx.x / slices;
    const int slice = blockIdx.x % slices;
    const int tid   = (int)threadIdx.x;
    const int chunk = n / slices;                 // 2048/8 = 256 == blockDim.x
    const int i     = slice * chunk + tid;        // row index in image0 / col index in image1

    const long long pitch = (long long)m + 1;     // 2049
    const float* sb = scores + (long long)b * (long long)(n + 1) * pitch;

    // Early prefetch of the dustbin-row element (global_prefetch_b8) so it
    // overlaps with the dependent gt0 load -> gather chain below.
    __builtin_prefetch(sb + (long long)n * pitch + i, 0, 1);

    float s = 0.0f;   // tp + tn partial
    float x = 0.0f;   // unmatched count partial

    // loss_tp contribution: scores[b, i, col] with col = (gt0<0 ? m : gt0)
    {
        int g   = gt0[(long long)b * n + i];
        int col = (g < 0) ? m : g;
        s += sb[(long long)i * pitch + col];
    }
    // loss_tn contribution: dustbin row where gt1 == -1
    {
        int g = gt1[(long long)b * m + i];        // n == m per reference
        if (g == -1) {
            s += sb[(long long)n * pitch + i];
            x += 1.0f;
        }
    }

    // Wave32 reduction via the matrix unit (EXEC is all-1s here: no divergence).
    float wsum = wave_sum_wmma(s);
    float wcnt = wave_sum_wmma(x);

    __shared__ float ss[8];
    __shared__ float sx[8];
    if ((tid & 31) == 0) {
        ss[tid >> 5] = wsum;
        sx[tid >> 5] = wcnt;
    }
    __syncthreads();
    if (tid == 0) {
        float S = 0.0f, X = 0.0f;
#pragma unroll
        for (int w = 0; w < 8; ++w) { S += ss[w]; X += sx[w]; }
        ws[(long long)blockIdx.x * 2 + 0] = S;
        ws[(long long)blockIdx.x * 2 + 1] = X;
    }
}

// Single wave: lane b (b < B) sums its slice partials, computes per-batch
// loss, then a 4-step shuffle tree averages over the 16 batches.
__global__ void __launch_bounds__(32)
superglue_finalize_kernel(const float* __restrict__ ws,
                          float* __restrict__ out,
                          int nbatch, int slices, int m) {
    const int lane = (int)threadIdx.x;
    float l = 0.0f;
    if (lane < nbatch) {
        float s = 0.0f, x = 0.0f;
        for (int sl = 0; sl < slices; ++sl) {
            s += ws[(long long)(lane * slices + sl) * 2 + 0];
            x += ws[(long long)(lane * slices + sl) * 2 + 1];
        }
        l = -s / (x + (float)m);
    }
    // lanes 16..31 contribute 0; reduce within the group of 16
    l += __shfl_xor(l, 8, 32);
    l += __shfl_xor(l, 4, 32);
    l += __shfl_xor(l, 2, 32);
    l += __shfl_xor(l, 1, 32);
    if (lane == 0) out[0] = l / (float)nbatch;
}

extern "C" void kernel_launch(void* const* d_in, const int* in_sizes, int n_in,
                              void* d_out, int out_size, void* d_ws, size_t ws_size,
                              hipStream_t stream) {
    const int*   gt0    = (const int*)d_in[0];     // [B, N] int32
    const int*   gt1    = (const int*)d_in[1];     // [B, M] int32
    const float* scores = (const float*)d_in[2];   // [B, N+1, M+1] f32
    float*       out    = (float*)d_out;           // scalar f32
    float*       ws     = (float*)d_ws;            // B*SLICES*2 floats = 1 KB

    const int B = 16, N = 2048, M = 2048;
    const int SLICES = 8;                          // 128 blocks total

    superglue_partial_kernel<<<dim3(B * SLICES), dim3(256), 0, stream>>>(
        gt0, gt1, scores, ws, N, M, SLICES);
    superglue_finalize_kernel<<<dim3(1), dim3(32), 0, stream>>>(
        ws, out, B, SLICES, M);
    (void)in_sizes; (void)n_in; (void)out_size; (void)ws_size;
}